// CopyMechModule_33827162423501
// MI455X (gfx1250) — compile-verified
//
#include <hip/hip_runtime.h>

#define B_  4
#define T_  512
#define S_  1024
#define H_  768
#define V_  50257
#define V4_ 12564   /* V_ = 4*V4_ + 1 */

typedef float v2f __attribute__((ext_vector_type(2)));
typedef float v8f __attribute__((ext_vector_type(8)));

#if defined(__AMDGCN__)
typedef __attribute__((address_space(1))) int gas_int;   // global ("__device__ int")
typedef __attribute__((address_space(3))) int las_int;   // LDS    ("__shared__ int")
#endif

// ---------------------------------------------------------------------------
// Kernel 1: fused row-dot products via V_WMMA_F32_16X16X4_F32 (one wave = 16 rows)
//   tiles [0, B*S/16)          : u[row] = dot(src[row,:], W[0:H])
//   tiles [B*S/16, +B*T/16)    : v[row] = dot(tgt[row,:], W[H:2H])
// B operand replicates the weight chunk across all 16 N columns, so every
// column of D holds the row-dot; lanes 0 and 16 extract column N==0.
// ---------------------------------------------------------------------------
__global__ __launch_bounds__(32)
void dots_uv_kernel(const float* __restrict__ src,
                    const float* __restrict__ tgt,
                    const float* __restrict__ W,
                    float* __restrict__ u,    // B*S floats
                    float* __restrict__ v)    // B*T floats
{
  const int tile = blockIdx.x;
  const float* mat;
  const float* w;
  float* out;
  int base;
  if (tile < (B_ * S_) / 16) {
    mat = src;  w = W;        base = tile * 16;                    out = u;
  } else {
    mat = tgt;  w = W + H_;   base = (tile - (B_ * S_) / 16) * 16; out = v;
  }

  const int lane = threadIdx.x & 31;
  const int half = lane >> 4;    // selects K pair {0,1} vs {2,3}
  const int m    = lane & 15;    // A-matrix row within tile
  const float* arow = mat + (size_t)(base + m) * H_;

  v8f acc = {};
  #pragma unroll 4
  for (int k0 = 0; k0 < H_; k0 += 4) {
    const int k = k0 + 2 * half;
    v2f a = *(const v2f*)(arow + k);          // A[m][k], A[m][k+1]
    v2f b;                                     // B[k][n] = w[k] for all n
    b.x = w[k];
    b.y = w[k + 1];
    acc = __builtin_amdgcn_wmma_f32_16x16x4_f32(
        /*neg_a=*/false, a, /*neg_b=*/false, b,
        /*c_mod=*/(short)0, acc, /*reuse_a=*/false, /*reuse_b=*/false);
  }

  if (m == 0) {   // column N==0 lives in lanes 0 (M=0..7) and 16 (M=8..15)
    #pragma unroll
    for (int r = 0; r < 8; ++r) out[base + half * 8 + r] = acc[r];
  }
}

// ---------------------------------------------------------------------------
// Kernel 2: p_gen[b,t] = sigmoid( attn[b,t,:]·u[b,:] + v[b,t] + bias )
// Same WMMA tiling: one wave per 16 (b,t) rows; T%16==0 so a tile never
// crosses a batch boundary.
// ---------------------------------------------------------------------------
__global__ __launch_bounds__(32)
void pgen_kernel(const float* __restrict__ attn,
                 const float* __restrict__ u,
                 const float* __restrict__ v,
                 const float* __restrict__ bias,
                 float* __restrict__ pgen)
{
  const int tile = blockIdx.x;
  const int base = tile * 16;          // row index over B*T
  const int b    = base / T_;
  const int lane = threadIdx.x & 31;
  const int half = lane >> 4;
  const int m    = lane & 15;
  const float* arow = attn + (size_t)(base + m) * S_;
  const float* ub   = u + b * S_;

  v8f acc = {};
  #pragma unroll 4
  for (int k0 = 0; k0 < S_; k0 += 4) {
    const int k = k0 + 2 * half;
    v2f a  = *(const v2f*)(arow + k);
    v2f bb = *(const v2f*)(ub + k);    // B[k][n] = u[b,k] for all n
    acc = __builtin_amdgcn_wmma_f32_16x16x4_f32(
        false, a, false, bb, (short)0, acc, false, false);
  }

  if (m == 0) {
    const float bs = bias[0];
    #pragma unroll
    for (int r = 0; r < 8; ++r) {
      const int row = base + half * 8 + r;
      const float x = acc[r] + v[row] + bs;
      pgen[row] = 1.0f / (1.0f + __expf(-x));
    }
  }
}

// ---------------------------------------------------------------------------
// Kernel 3: copy-logits scatter. One workgroup per (b,t) row.
// Full vocab row (201 KB) lives in CDNA5's 320 KB WGP LDS: zero it, scatter
// S=1024 attention weights with ds_add_f32 (one per thread), then stream the
// row to HBM. The streaming phase uses GLOBAL_STORE_ASYNC_FROM_LDS_B32
// (ASYNCcnt path): data moves LDS -> memory directly, never touching VGPRs;
// b32 because rows are only 4B-aligned (V odd).
// This fuses the zero-fill with the scatter: exactly one 412 MB write pass.
// ---------------------------------------------------------------------------
__global__ __launch_bounds__(1024)
void scatter_kernel(const int* __restrict__ ids,
                    const float* __restrict__ attn,
                    float* __restrict__ logits)
{
  extern __shared__ float row[];           // V_ floats = 201 KB
  const int bt  = blockIdx.x;              // 0 .. B*T-1
  const int b   = bt / T_;
  const int tid = threadIdx.x;

  // zero vocab row in LDS (vectorized; tail element separately)
  float4* row4 = (float4*)row;
  const float4 z4 = make_float4(0.f, 0.f, 0.f, 0.f);
  for (int i = tid; i < V4_; i += 1024) row4[i] = z4;
  if (tid == 0) row[V_ - 1] = 0.0f;
  __syncthreads();

  // scatter: S_ == 1024 == blockDim, one ds_add_f32 per thread
  {
    const float wgt = attn[(size_t)bt * S_ + tid];
    const int   id  = ids[b * S_ + tid];
    atomicAdd(&row[id], wgt);
  }
  __syncthreads();

  // stream row out
  float* out = logits + (size_t)bt * V_;
#if __has_builtin(__builtin_amdgcn_global_store_async_from_lds_b32)
  for (int i = tid; i < V_; i += 1024) {
    __builtin_amdgcn_global_store_async_from_lds_b32(
        (gas_int*)(out + i), (las_int*)(row + i), /*imm_offset=*/0, /*cpol=*/0);
  }
  #if __has_builtin(__builtin_amdgcn_s_wait_asynccnt)
  __builtin_amdgcn_s_wait_asynccnt(0);   // hygiene; S_ENDPGM also waits idle
  #endif
#else
  for (int i = tid; i < V_; i += 1024)
    __builtin_nontemporal_store(row[i], out + i);
#endif
}

// ---------------------------------------------------------------------------
extern "C" void kernel_launch(void* const* d_in, const int* in_sizes, int n_in,
                              void* d_out, int out_size, void* d_ws, size_t ws_size,
                              hipStream_t stream) {
  const int*   ids  = (const int*)  d_in[0];   // [B,S]
  const float* attn = (const float*)d_in[1];   // [B,T,S]
  const float* src  = (const float*)d_in[2];   // [B,S,H]
  const float* tgt  = (const float*)d_in[3];   // [B,T,H]
  const float* W    = (const float*)d_in[4];   // [2H,1]
  const float* bias = (const float*)d_in[5];   // [1]

  float* pgen   = (float*)d_out;                       // [B,T,1] -> B*T floats
  float* logits = (float*)d_out + (size_t)B_ * T_;     // [B,T,V]

  float* u = (float*)d_ws;        // B*S floats
  float* v = u + B_ * S_;         // B*T floats

  const int lds_bytes = ((V_ * 4 + 255) / 256) * 256;  // 201,216 B of the 320 KB WGP LDS
  (void)hipFuncSetAttribute((const void*)scatter_kernel,
                            hipFuncAttributeMaxDynamicSharedMemorySize, lds_bytes);

  dots_uv_kernel<<<(B_ * S_ + B_ * T_) / 16, 32, 0, stream>>>(src, tgt, W, u, v);
  pgen_kernel<<<(B_ * T_) / 16, 32, 0, stream>>>(attn, u, v, bias, pgen);
  scatter_kernel<<<B_ * T_, 1024, lds_bytes, stream>>>(ids, attn, logits);
}